// TemporalAggregationBlock_29944511988247
// MI455X (gfx1250) — compile-verified
//
#include <hip/hip_runtime.h>
#include <math.h>

typedef float v2f __attribute__((ext_vector_type(2)));
typedef float v8f __attribute__((ext_vector_type(8)));

#define LL     197
#define NTOT   128
#define DD     768
#define TT     16
#define PP     196
#define NGROUP 8
#define NTOK   (LL * NTOT)      /* 25216 tokens */
#define STR    772              /* padded LDS row stride (768 + 4) */
#define AFF_SCALE 0.03608439182435161f  /* 1/sqrt(768) */

/* ---------------- Kernel 1: per-token layernorm stats (mu, rstd) ------------- */
__global__ void ln_stats_kernel(const float* __restrict__ x,
                                float* __restrict__ mu,
                                float* __restrict__ rstd)
{
    int wave = threadIdx.x >> 5;
    int lane = threadIdx.x & 31;
    int tok  = blockIdx.x * 8 + wave;
    if (tok >= NTOK) return;
    const float* row = x + (size_t)tok * DD;
    float s = 0.f, sq = 0.f;
    for (int i = lane; i < DD; i += 32) {
        float v = row[i];
        s += v; sq += v * v;
    }
    #pragma unroll
    for (int m = 16; m >= 1; m >>= 1) {
        s  += __shfl_xor(s,  m);
        sq += __shfl_xor(sq, m);
    }
    if (lane == 0) {
        float mean = s * (1.f / DD);
        float var  = sq * (1.f / DD) - mean * mean;
        mu[tok]   = mean;
        rstd[tok] = rsqrtf(var + 1e-5f);
    }
}

/* ---------------- Kernel 2: banded sigmoid scores G_p^b (16x16 each) ---------
 * grid = NGROUP*PP blocks, 256 threads (8 waves).
 * S = C (16x768) @ X^T (768x16) via v_wmma_f32_16x16x4_f32, K split over waves.
 */
__global__ void scores_kernel(const float* __restrict__ x,
                              const float* __restrict__ gamma,
                              const float* __restrict__ beta,
                              const float* __restrict__ mu,
                              const float* __restrict__ rstd,
                              float* __restrict__ gout)
{
    extern __shared__ float sh[];
    float* Cs = sh;               /* 16 * STR */
    float* Xs = sh + 16 * STR;    /* 16 * STR */
    float* Sp = sh + 32 * STR;    /* 8 * 256 partials */

    int b  = blockIdx.x / PP;
    int p  = blockIdx.x - b * PP;
    int nb = b * TT;
    int tid = threadIdx.x;
    int row = tid >> 4;           /* 0..15 : sample row within group */
    int l16 = tid & 15;

    /* Load + layernorm CLS row and patch row into LDS (float4 loads). */
    {
        int tokC = nb + row;                      /* l = 0 (CLS)      */
        int tokX = (1 + p) * NTOT + nb + row;     /* l = 1+p (patch)  */
        float muC = mu[tokC], rsC = rstd[tokC];
        float muX = mu[tokX], rsX = rstd[tokX];
        const float4* xc  = (const float4*)(x + (size_t)tokC * DD);
        const float4* xx  = (const float4*)(x + (size_t)tokX * DD);
        const float4* g4p = (const float4*)gamma;
        const float4* b4p = (const float4*)beta;
        for (int j = l16; j < DD / 4; j += 16) {
            float4 cv = xc[j], xv = xx[j];
            float4 gv = g4p[j], bv = b4p[j];
            float* cdst = Cs + row * STR + j * 4;
            float* xdst = Xs + row * STR + j * 4;
            cdst[0] = (cv.x - muC) * rsC * gv.x + bv.x;
            cdst[1] = (cv.y - muC) * rsC * gv.y + bv.y;
            cdst[2] = (cv.z - muC) * rsC * gv.z + bv.z;
            cdst[3] = (cv.w - muC) * rsC * gv.w + bv.w;
            xdst[0] = (xv.x - muX) * rsX * gv.x + bv.x;
            xdst[1] = (xv.y - muX) * rsX * gv.y + bv.y;
            xdst[2] = (xv.z - muX) * rsX * gv.z + bv.z;
            xdst[3] = (xv.w - muX) * rsX * gv.w + bv.w;
        }
    }
    __syncthreads();

    int wave = tid >> 5, lane = tid & 31;
    int arow = lane & 15;          /* A: row m / B: col n' */
    int kofs = 2 * (lane >> 4);    /* K pair select within chunk of 4 */
    const float* crow = Cs + arow * STR + kofs;
    const float* xrow = Xs + arow * STR + kofs;
    int kbase = wave * 96;         /* each wave covers 96 of K=768 */

    v8f acc = {};
    #pragma unroll
    for (int kc = 0; kc < 24; ++kc) {
        int k0 = kbase + kc * 4;
        v2f av = *(const v2f*)(crow + k0);   /* b64 DS load, conflict-free */
        v2f bv = *(const v2f*)(xrow + k0);
        acc = __builtin_amdgcn_wmma_f32_16x16x4_f32(
            false, av, false, bv, (short)0, acc, false, false);
    }

    /* Cross-wave K reduction of the 16x16 partials. */
    #pragma unroll
    for (int j = 0; j < 8; ++j) {
        int m = j + 8 * (lane >> 4);
        Sp[wave * 256 + m * 16 + (lane & 15)] = acc[j];
    }
    __syncthreads();

    float s = 0.f;
    #pragma unroll
    for (int w = 0; w < 8; ++w) s += Sp[w * 256 + tid];

    int nc = tid >> 4;             /* CLS sample index (S row)      */
    int nn = tid & 15;             /* neighbor sample index (S col) */
    float g = 0.f;
    int dlt = nn - nc;
    if (dlt >= -2 && dlt <= 2) {
        float aff = s * AFF_SCALE;
        g = 1.f / (1.f + __expf(-aff)) - 0.5f;
    }
    /* store as G[n'][n] so stage-2 B operand reads row = n', col = n */
    gout[((size_t)blockIdx.x) * 256 + nn * 16 + nc] = g;
}

/* ---------------- Kernel 3: feat tile = sum_p X_p^T(16x16) @ G_p(16x16) ------
 * grid = NGROUP * 48 blocks (one 16-wide d-tile per block), 8 waves split p.
 */
__global__ void feat_kernel(const float* __restrict__ x,
                            const float* __restrict__ gamma,
                            const float* __restrict__ beta,
                            const float* __restrict__ mu,
                            const float* __restrict__ rstd,
                            const float* __restrict__ gin,
                            float* __restrict__ out)
{
    __shared__ float Sp[8 * 256];

    int b    = blockIdx.x / 48;
    int dt   = blockIdx.x - b * 48;
    int d0   = dt * 16;
    int nb   = b * TT;
    int wave = threadIdx.x >> 5, lane = threadIdx.x & 31;
    int m     = lane & 15;        /* A: d row within tile; B: output col n */
    int rsel2 = 2 * (lane >> 4);  /* K pair select */
    int dcol  = d0 + m;
    float gam = gamma[dcol], bet = beta[dcol];

    v8f acc = {};
    for (int p = wave; p < PP; p += 8) {
        const float* gp = gin + ((size_t)(b * PP + p)) * 256;
        int    tokbase  = (1 + p) * NTOT + nb;
        size_t xbase    = (size_t)tokbase * DD;
        if (p + 8 < PP) {   /* wave-uniform prefetch of next patch tile */
            __builtin_prefetch(x + xbase + (size_t)8 * NTOT * DD + dcol, 0, 0);
        }
        #pragma unroll
        for (int kk = 0; kk < 4; ++kk) {
            int r0 = kk * 4 + rsel2;   /* neighbor rows for this K pair */
            int r1 = r0 + 1;
            float x0 = x[xbase + (size_t)r0 * DD + dcol];
            float x1 = x[xbase + (size_t)r1 * DD + dcol];
            v2f av = { (x0 - mu[tokbase + r0]) * rstd[tokbase + r0] * gam + bet,
                       (x1 - mu[tokbase + r1]) * rstd[tokbase + r1] * gam + bet };
            v2f bv = { gp[r0 * 16 + m], gp[r1 * 16 + m] };
            acc = __builtin_amdgcn_wmma_f32_16x16x4_f32(
                false, av, false, bv, (short)0, acc, false, false);
        }
    }

    /* Cross-wave p reduction. */
    #pragma unroll
    for (int j = 0; j < 8; ++j) {
        int mm = j + 8 * (lane >> 4);
        Sp[wave * 256 + mm * 16 + (lane & 15)] = acc[j];
    }
    __syncthreads();

    float s = 0.f;
    #pragma unroll
    for (int w = 0; w < 8; ++w) s += Sp[w * 256 + threadIdx.x];

    int md = threadIdx.x >> 4;    /* d within tile  */
    int n  = threadIdx.x & 15;    /* sample in group */
    out[(size_t)(nb + n) * DD + d0 + md] = s;
}

extern "C" void kernel_launch(void* const* d_in, const int* in_sizes, int n_in,
                              void* d_out, int out_size, void* d_ws, size_t ws_size,
                              hipStream_t stream)
{
    (void)in_sizes; (void)n_in; (void)out_size; (void)ws_size;
    const float* x     = (const float*)d_in[0];
    const float* gamma = (const float*)d_in[1];
    const float* beta  = (const float*)d_in[2];
    /* d_in[3] is T == 16, baked into the kernels. */

    float* ws   = (float*)d_ws;
    float* mu   = ws;                 /* NTOK floats            */
    float* rstd = ws + NTOK;          /* NTOK floats            */
    float* g    = ws + 2 * NTOK;      /* NGROUP*PP*256 floats   */

    ln_stats_kernel<<<NTOK / 8, 256, 0, stream>>>(x, mu, rstd);

    size_t shmem = (size_t)(32 * STR + 8 * 256) * sizeof(float);
    scores_kernel<<<NGROUP * PP, 256, shmem, stream>>>(x, gamma, beta, mu, rstd, g);

    feat_kernel<<<NGROUP * 48, 256, 0, stream>>>(x, gamma, beta, mu, rstd, g,
                                                 (float*)d_out);
}